// MMD_loss_38646115730031
// MI455X (gfx1250) — compile-verified
//
#include <hip/hip_runtime.h>
#include <hip/hip_bf16.h>

typedef __attribute__((ext_vector_type(2))) float v2f;
typedef __attribute__((ext_vector_type(8))) float v8f;

#define NS 4096
#define NT 4096
#define DD 256
#define KC 32               // K chunk staged in LDS
#define BT 128              // block tile edge
#define LSTR (KC + 4)       // padded LDS row stride (floats)

// ---------------- helper kernels ----------------

__global__ void mmd_init_acc(float* __restrict__ acc) {
    if (threadIdx.x < 4) acc[threadIdx.x] = 0.0f;
}

// one 256-thread block per row: out[row] = sum_k X[row][k]^2
__global__ __launch_bounds__(256) void mmd_row_norms(const float* __restrict__ X,
                                                     float* __restrict__ out) {
    __shared__ float red[256];
    const int row = blockIdx.x;
    const float v = X[row * DD + threadIdx.x];
    red[threadIdx.x] = v * v;
    __syncthreads();
    for (int off = 128; off > 0; off >>= 1) {
        if (threadIdx.x < off) red[threadIdx.x] += red[threadIdx.x + off];
        __syncthreads();
    }
    if (threadIdx.x == 0) out[row] = red[0];
}

// w[i] = class_weight[label[i]]; nacc += sum(w)
__global__ __launch_bounds__(256) void mmd_weights(const int* __restrict__ label,
                                                   const float* __restrict__ cw,
                                                   float* __restrict__ w,
                                                   float* __restrict__ nacc) {
    __shared__ float red[256];
    const int i = blockIdx.x * 256 + threadIdx.x;
    const float wi = cw[label[i]];
    w[i] = wi;
    red[threadIdx.x] = wi;
    __syncthreads();
    for (int off = 128; off > 0; off >>= 1) {
        if (threadIdx.x < off) red[threadIdx.x] += red[threadIdx.x + off];
        __syncthreads();
    }
    if (threadIdx.x == 0) atomicAdd(nacc, red[0]);
}

// ---------------- main fused Gram + exp + weighted-reduce kernel ----------------
// blockIdx.z: 0 -> sum w_i w_j Kss ; 1 -> sum Ktt ; 2 -> sum w_i Kst
__global__ __launch_bounds__(256) void mmd_gram_fused(
    const float* __restrict__ S, const float* __restrict__ T,
    const float* __restrict__ x2s, const float* __restrict__ x2t,
    const float* __restrict__ w, float* __restrict__ acc) {

    const int mode = blockIdx.z;
    const float* X;  const float* Y;
    const float* x2X; const float* x2Y;
    if (mode == 0)      { X = S; Y = S; x2X = x2s; x2Y = x2s; }
    else if (mode == 1) { X = T; Y = T; x2X = x2t; x2Y = x2t; }
    else                { X = S; Y = T; x2X = x2s; x2Y = x2t; }

    const int i0 = blockIdx.x * BT;
    const int j0 = blockIdx.y * BT;

    __shared__ float Xs[BT][LSTR];
    __shared__ float Ys[BT][LSTR];

    const int tid  = threadIdx.x;
    const int wave = tid >> 5;      // 0..7
    const int lane = tid & 31;
    const int hl   = lane >> 4;     // half-wave 0/1
    const int l16  = lane & 15;

    v8f accf[8];
#pragma unroll
    for (int t = 0; t < 8; ++t)
#pragma unroll
        for (int e = 0; e < 8; ++e) accf[t][e] = 0.0f;

    for (int k0 = 0; k0 < DD; k0 += KC) {
        // stage 128x32 tiles of X and Y into LDS (float4 per thread x4)
#pragma unroll
        for (int it = 0; it < 4; ++it) {
            const int idx = tid + it * 256;   // 0..1023
            const int r   = idx >> 3;         // 0..127
            const int c4  = idx & 7;          // 0..7 (float4 column)
            const float4 vx = *(const float4*)&X[(size_t)(i0 + r) * DD + k0 + c4 * 4];
            const float4 vy = *(const float4*)&Y[(size_t)(j0 + r) * DD + k0 + c4 * 4];
            *(float4*)&Xs[r][c4 * 4] = vx;
            *(float4*)&Ys[r][c4 * 4] = vy;
        }
        // prefetch next K chunk into cache (global_prefetch_b8)
        if (k0 + KC < DD) {
            const int r = tid >> 1;
            __builtin_prefetch(&X[(size_t)(i0 + r) * DD + k0 + KC], 0, 3);
            __builtin_prefetch(&Y[(size_t)(j0 + r) * DD + k0 + KC], 0, 3);
        }
        __syncthreads();

        // GEMM: wave owns 16 rows (16*wave..), 8 column tiles of 16
#pragma unroll
        for (int kk = 0; kk < KC; kk += 4) {
            v2f a;
            a[0] = Xs[16 * wave + l16][kk + 2 * hl + 0];
            a[1] = Xs[16 * wave + l16][kk + 2 * hl + 1];
#pragma unroll
            for (int t = 0; t < 8; ++t) {
                v2f b;
                b[0] = Ys[16 * t + l16][kk + 2 * hl + 0];
                b[1] = Ys[16 * t + l16][kk + 2 * hl + 1];
                accf[t] = __builtin_amdgcn_wmma_f32_16x16x4_f32(
                    /*neg_a=*/false, a, /*neg_b=*/false, b,
                    /*c_mod=*/(short)0, accf[t],
                    /*reuse_a=*/false, /*reuse_b=*/false);
            }
        }
        __syncthreads();
    }

    // epilogue: accf[t][r] = dot(X[i], Y[j]) with
    //   i = i0 + 16*wave + 8*hl + r ; j = j0 + 16*t + l16
    const int ibase = i0 + 16 * wave + 8 * hl;
    float x2i[8], wi8[8];
#pragma unroll
    for (int r = 0; r < 8; ++r) {
        x2i[r] = x2X[ibase + r];
        wi8[r] = (mode == 1) ? 1.0f : w[ibase + r];
    }

    float lsum = 0.0f;
#pragma unroll
    for (int t = 0; t < 8; ++t) {
        const int j   = j0 + 16 * t + l16;
        const float y2 = x2Y[j];
        const float wj = (mode == 0) ? w[j] : 1.0f;
#pragma unroll
        for (int r = 0; r < 8; ++r) {
            float sq = x2i[r] + y2 - 2.0f * accf[t][r];
            sq = fmaxf(sq, 0.0f);
            const float kv = __expf(-0.5f * sq);   // SIGMA = 1
            lsum += wi8[r] * wj * kv;
        }
    }

    // block reduction (reuse Xs storage) + atomic accumulate
    __syncthreads();
    float* red = &Xs[0][0];
    red[tid] = lsum;
    __syncthreads();
    for (int off = 128; off > 0; off >>= 1) {
        if (tid < off) red[tid] += red[tid + off];
        __syncthreads();
    }
    if (tid == 0) atomicAdd(&acc[mode], red[0]);
}

__global__ void mmd_finalize(const float* __restrict__ acc, float* __restrict__ out) {
    const float first = acc[0], sec = acc[1], third = acc[2], n = acc[3];
    const float nt = (float)NT;
    out[0] = first / (n * n) + sec / (nt * nt) - 2.0f * third / (n * nt);
}

// ---------------- launch ----------------

extern "C" void kernel_launch(void* const* d_in, const int* in_sizes, int n_in,
                              void* d_out, int out_size, void* d_ws, size_t ws_size,
                              hipStream_t stream) {
    (void)in_sizes; (void)n_in; (void)out_size; (void)ws_size;

    const float* src  = (const float*)d_in[0];   // [4096, 256]
    const int*   lbl  = (const int*)  d_in[1];   // [4096]
    const float* tgt  = (const float*)d_in[2];   // [4096, 256]
    const float* cw   = (const float*)d_in[3];   // [10]
    float* out = (float*)d_out;

    float* acc = (float*)d_ws;        // [0..3]: first, sec, third, n
    float* x2s = acc + 16;            // [NS]
    float* x2t = x2s + NS;            // [NT]
    float* wv  = x2t + NT;            // [NS]

    mmd_init_acc<<<1, 4, 0, stream>>>(acc);
    mmd_row_norms<<<NS, 256, 0, stream>>>(src, x2s);
    mmd_row_norms<<<NT, 256, 0, stream>>>(tgt, x2t);
    mmd_weights<<<NS / 256, 256, 0, stream>>>(lbl, cw, wv, acc + 3);

    dim3 grid(NS / BT, NT / BT, 3);
    mmd_gram_fused<<<grid, 256, 0, stream>>>(src, tgt, x2s, x2t, wv, acc);

    mmd_finalize<<<1, 1, 0, stream>>>(acc, out);
}